// _HeteroGNNModule_57363583205826
// MI455X (gfx1250) — compile-verified
//
#include <hip/hip_runtime.h>

// Problem constants (match reference)
constexpr int NDRUG = 50000;
constexpr int NDIS  = 25000;
constexpr int DIMD  = 256;
constexpr int NEDGE = 300000;

typedef __attribute__((ext_vector_type(2))) float v2f;
typedef __attribute__((ext_vector_type(4))) float v4f;
typedef __attribute__((ext_vector_type(8))) float v8f;

// ---------------------------------------------------------------- zero fill
__global__ void zero_kernel(float* __restrict__ p, long n) {
    long i = (long)blockIdx.x * blockDim.x + threadIdx.x;
    long stride = (long)gridDim.x * blockDim.x;
    for (; i < n; i += stride) p[i] = 0.0f;
}

// ------------------------------------------------- edge scatter (segment sum)
// One wave32 per edge: each lane atomically adds 8 floats of the 256-wide row.
// Destination buffers are L2-resident (<=51MB) so these atomics stay on-chip.
__global__ void scatter_kernel(const float* __restrict__ xsrc,
                               const int* __restrict__ src,
                               const int* __restrict__ dst,
                               float* __restrict__ agg,
                               float* __restrict__ cnt, int nE) {
    int edge = blockIdx.x * (blockDim.x >> 5) + (threadIdx.x >> 5);
    int lane = threadIdx.x & 31;
    if (edge >= nE) return;
    int s = src[edge];
    int d = dst[edge];
    const float* xr = xsrc + (long)s * DIMD;
    float* ar = agg + (long)d * DIMD;
#pragma unroll
    for (int i = 0; i < DIMD; i += 32)
        atomicAdd(&ar[i + lane], xr[i + lane]);
    if (lane == 0) atomicAdd(&cnt[d], 1.0f);
}

// ---------------------------------------------------------------- mean divide
__global__ void mean_kernel(float* __restrict__ agg,
                            const float* __restrict__ cnt, int nrows) {
    long idx = (long)blockIdx.x * blockDim.x + threadIdx.x;
    long n = (long)nrows * DIMD;
    if (idx >= n) return;
    float c = cnt[idx >> 8];           // DIMD == 256
    agg[idx] = agg[idx] / fmaxf(c, 1.0f);
}

// ------------------------------------------------------------ fused WMMA GEMM
// Y[M,256] = act( scale * ( sum_{p<npairs} Xp @ Wp  + bias1 + bias2 ) )
// Block: 256 threads (8 waves), output tile 128(M) x 128(N).
// Wave grid 2(M) x 4(N): each wave owns 64x32 => 4x2 fragments of 16x16.
// V_WMMA_F32_16X16X4_F32 (exact f32 on the matrix pipe).
// W is staged TRANSPOSED in LDS so both A and B fragments are single aligned
// ds_load_b64; global loads for stage s+1 are register-double-buffered under
// the 64-WMMA compute of stage s.
__launch_bounds__(256)
__global__ void gemm_kernel(const float* __restrict__ X0, const float* __restrict__ W0,
                            const float* __restrict__ X1, const float* __restrict__ W1,
                            const float* __restrict__ X2, const float* __restrict__ W2,
                            const float* __restrict__ X3, const float* __restrict__ W3,
                            const float* __restrict__ bias1, const float* __restrict__ bias2,
                            float* __restrict__ Y, int M, float scale, int relu, int npairs) {
    // pad 36: 16-lane fragment reads (stride 36) hit distinct banks; 8B aligned
    __shared__ float ldsX[128][36];    // [m][k]
    __shared__ float ldsWt[128][36];   // [n][k]  (transposed)

    const int tid  = threadIdx.x;
    const int lane = tid & 31;
    const int wave = tid >> 5;
    const int hf   = lane >> 4;        // lane half
    const int l15  = lane & 15;
    const int wm   = (wave & 1) * 64;
    const int wn   = (wave >> 1) * 32;
    const int mbase = blockIdx.x * 128;
    const int nbase = blockIdx.y * 128;

    // W staging: each thread owns one output column and 16 consecutive k's.
    const int wcol = tid & 127;
    const int wkg  = (tid >> 7) * 16;

    v8f acc[4][2];
#pragma unroll
    for (int mi = 0; mi < 4; ++mi)
#pragma unroll
        for (int ni = 0; ni < 2; ++ni)
            acc[mi][ni] = v8f{};

    const int nstage = npairs * 8;     // 8 K-slabs of 32 per pair (K = 256)

    v4f  xreg[4];
    float wreg[16];

    auto stage_load = [&](int s) {
        int p  = s >> 3;
        int kb = (s & 7) << 5;
        const float* X = X0;
        const float* W = W0;
        if (p == 1)      { X = X1; W = W1; }
        else if (p == 2) { X = X2; W = W2; }
        else if (p == 3) { X = X3; W = W3; }
#pragma unroll
        for (int i = 0; i < 4; ++i) {
            int f  = tid + i * 256;
            int r  = f >> 3;
            int c4 = (f & 7) * 4;
            int grow = mbase + r;
            xreg[i] = v4f{};
            if (grow < M) xreg[i] = *(const v4f*)(X + (long)grow * DIMD + kb + c4);
        }
#pragma unroll
        for (int j = 0; j < 16; ++j)   // coalesced: lanes sweep consecutive cols
            wreg[j] = W[(long)(kb + wkg + j) * DIMD + nbase + wcol];
    };

    stage_load(0);

    for (int s = 0; s < nstage; ++s) {
        __syncthreads();
        // commit prefetched registers to LDS
#pragma unroll
        for (int i = 0; i < 4; ++i) {
            int f  = tid + i * 256;
            int r  = f >> 3;
            int c4 = (f & 7) * 4;
            ldsX[r][c4 + 0] = xreg[i].x; ldsX[r][c4 + 1] = xreg[i].y;
            ldsX[r][c4 + 2] = xreg[i].z; ldsX[r][c4 + 3] = xreg[i].w;
        }
#pragma unroll
        for (int j = 0; j < 16; j += 4) {
            v4f v; v.x = wreg[j]; v.y = wreg[j + 1]; v.z = wreg[j + 2]; v.w = wreg[j + 3];
            *(v4f*)&ldsWt[wcol][wkg + j] = v;   // 16B aligned (36,wkg,j mult. of 4)
        }
        __syncthreads();

        // issue next stage's global loads under this stage's compute
        if (s + 1 < nstage) stage_load(s + 1);

#pragma unroll
        for (int kk = 0; kk < 32; kk += 4) {
            // B fragments (4x16): single aligned b64 per fragment
            v2f bfr[2];
#pragma unroll
            for (int ni = 0; ni < 2; ++ni) {
                int col = wn + ni * 16 + l15;
                bfr[ni].x = ldsWt[col][kk + 2 * hf];
                bfr[ni].y = ldsWt[col][kk + 2 * hf + 1];
            }
#pragma unroll
            for (int mi = 0; mi < 4; ++mi) {
                // A fragment (16x4): lanes 0-15 K0/K1, lanes 16-31 K2/K3
                int row = wm + mi * 16 + l15;
                v2f a;
                a.x = ldsX[row][kk + 2 * hf];
                a.y = ldsX[row][kk + 2 * hf + 1];
#pragma unroll
                for (int ni = 0; ni < 2; ++ni) {
                    acc[mi][ni] = __builtin_amdgcn_wmma_f32_16x16x4_f32(
                        false, a, false, bfr[ni], (short)0, acc[mi][ni],
                        false, false);
                }
            }
        }
    }

    // epilogue: bias + scale + relu; C/D layout: VGPR r -> M=r (+8 hi half)
#pragma unroll
    for (int ni = 0; ni < 2; ++ni) {
        int col = nbase + wn + ni * 16 + l15;
        float bsum = bias1[col] + (bias2 ? bias2[col] : 0.0f);
#pragma unroll
        for (int mi = 0; mi < 4; ++mi) {
#pragma unroll
            for (int r = 0; r < 8; ++r) {
                int row = mbase + wm + mi * 16 + (hf ? 8 : 0) + r;
                if (row < M) {
                    float v = scale * (acc[mi][ni][r] + bsum);
                    if (relu) v = fmaxf(v, 0.0f);
                    Y[(long)row * DIMD + col] = v;
                }
            }
        }
    }
}

// ------------------------------------------------------------------- driver
extern "C" void kernel_launch(void* const* d_in, const int* in_sizes, int n_in,
                              void* d_out, int out_size, void* d_ws, size_t ws_size,
                              hipStream_t stream) {
    const float* emb_drug = (const float*)d_in[0];
    const float* emb_dis  = (const float*)d_in[1];
    const float* Wl = (const float*)d_in[2];   // (2,3,256,256)
    const float* Wr = (const float*)d_in[3];   // (2,3,256,256)
    const float* bb = (const float*)d_in[4];   // (2,3,256)
    const int* treats_src  = (const int*)d_in[5];
    const int* treats_dst  = (const int*)d_in[6];
    const int* treated_src = (const int*)d_in[7];
    const int* treated_dst = (const int*)d_in[8];
    const int* inter_src   = (const int*)d_in[9];
    const int* inter_dst   = (const int*)d_in[10];

    float* ws = (float*)d_ws;
    float* agg_dis = ws;                                   // 25000*256
    float* agg_d1  = agg_dis + (long)NDIS * DIMD;          // 50000*256
    float* agg_d2  = agg_d1  + (long)NDRUG * DIMD;         // 50000*256
    float* cnt_dis = agg_d2  + (long)NDRUG * DIMD;         // 25000
    float* cnt_d1  = cnt_dis + NDIS;                       // 50000
    float* cnt_d2  = cnt_d1  + NDRUG;                      // 50000
    float* xdr_buf = cnt_d2  + NDRUG;                      // 50000*256
    float* xdi_buf = xdr_buf + (long)NDRUG * DIMD;         // 25000*256

    float* out_dr_final = (float*)d_out;
    float* out_di_final = (float*)d_out + (long)NDRUG * DIMD;

    const float* x_dr = emb_drug;
    const float* x_di = emb_dis;

    const int sc_blocks = (NEDGE + 7) / 8;  // 8 waves per 256-thread block

    for (int l = 0; l < 2; ++l) {
        float* o_dr = (l == 1) ? out_dr_final : xdr_buf;
        float* o_di = (l == 1) ? out_di_final : xdi_buf;

        zero_kernel<<<2048, 256, 0, stream>>>(agg_dis, (long)NDIS * DIMD);
        zero_kernel<<<2048, 256, 0, stream>>>(agg_d1,  (long)NDRUG * DIMD);
        zero_kernel<<<2048, 256, 0, stream>>>(agg_d2,  (long)NDRUG * DIMD);
        zero_kernel<<<64,   256, 0, stream>>>(cnt_dis, NDIS);
        zero_kernel<<<64,   256, 0, stream>>>(cnt_d1,  NDRUG);
        zero_kernel<<<64,   256, 0, stream>>>(cnt_d2,  NDRUG);

        scatter_kernel<<<sc_blocks, 256, 0, stream>>>(x_dr, treats_src,  treats_dst,  agg_dis, cnt_dis, NEDGE);
        scatter_kernel<<<sc_blocks, 256, 0, stream>>>(x_di, treated_src, treated_dst, agg_d1,  cnt_d1,  NEDGE);
        scatter_kernel<<<sc_blocks, 256, 0, stream>>>(x_dr, inter_src,   inter_dst,   agg_d2,  cnt_d2,  NEDGE);

        mean_kernel<<<((long)NDIS * DIMD + 255) / 256, 256, 0, stream>>>(agg_dis, cnt_dis, NDIS);
        mean_kernel<<<((long)NDRUG * DIMD + 255) / 256, 256, 0, stream>>>(agg_d1, cnt_d1, NDRUG);
        mean_kernel<<<((long)NDRUG * DIMD + 255) / 256, 256, 0, stream>>>(agg_d2, cnt_d2, NDRUG);

        const float* Wl0 = Wl + ((long)l * 3 + 0) * DIMD * DIMD;
        const float* Wl1 = Wl + ((long)l * 3 + 1) * DIMD * DIMD;
        const float* Wl2 = Wl + ((long)l * 3 + 2) * DIMD * DIMD;
        const float* Wr0 = Wr + ((long)l * 3 + 0) * DIMD * DIMD;
        const float* Wr1 = Wr + ((long)l * 3 + 1) * DIMD * DIMD;
        const float* Wr2 = Wr + ((long)l * 3 + 2) * DIMD * DIMD;
        const float* b0  = bb + ((long)l * 3 + 0) * DIMD;
        const float* b1  = bb + ((long)l * 3 + 1) * DIMD;
        const float* b2  = bb + ((long)l * 3 + 2) * DIMD;

        // disease: relu( mean@Wl0 + b0 + x_dis@Wr0 )
        dim3 gdis((NDIS + 127) / 128, 2);
        gemm_kernel<<<gdis, 256, 0, stream>>>(agg_dis, Wl0, x_di, Wr0,
                                              nullptr, nullptr, nullptr, nullptr,
                                              b0, nullptr, o_di, NDIS, 1.0f, 1, 2);
        // drug: relu( 0.5*( mean1@Wl1 + b1 + x@Wr1 + mean2@Wl2 + b2 + x@Wr2 ) )
        dim3 gdr((NDRUG + 127) / 128, 2);
        gemm_kernel<<<gdr, 256, 0, stream>>>(agg_d1, Wl1, x_dr, Wr1,
                                             agg_d2, Wl2, x_dr, Wr2,
                                             b1, b2, o_dr, NDRUG, 0.5f, 1, 4);

        x_dr = o_dr;
        x_di = o_di;
    }
    (void)in_sizes; (void)n_in; (void)out_size; (void)ws_size;
}